// GCNEncoder_24386824307021
// MI455X (gfx1250) — compile-verified
//
#include <hip/hip_runtime.h>

typedef float v2f __attribute__((ext_vector_type(2)));
typedef float v8f __attribute__((ext_vector_type(8)));

#define NNODES 100000
#define NEDGES 1600000
#define NGRAPH 64
#define DIN    128
#define DH     128
#define DOUT   64

// ---------------------------------------------------------------- utilities
__global__ void fill_kernel(float* __restrict__ p, float v, int n) {
    int i = blockIdx.x * blockDim.x + threadIdx.x;
    if (i < n) p[i] = v;
}

// deg[dst] += 1 over the E real edges (deg pre-filled with 1.0 for self loops)
__global__ void degree_kernel(const int* __restrict__ dst, float* __restrict__ deg) {
    int e = blockIdx.x * blockDim.x + threadIdx.x;
    if (e < NEDGES) atomicAdd(&deg[dst[e]], 1.0f);
}

// deg -> deg^{-1/2} in place
__global__ void dinv_kernel(float* __restrict__ deg) {
    int i = blockIdx.x * blockDim.x + threadIdx.x;
    if (i < NNODES) {
        float d = deg[i];
        deg[i] = (d > 0.0f) ? rsqrtf(d) : 0.0f;
    }
}

// ---------------------------------------------------------------- WMMA GEMM
// C[nrows, dout] = A[nrows, din] @ W[din, dout]
// blockDim = 256 (8 wave32). Each wave owns one 16x16 output tile.
// wavesPerRow = dout/16 col-tiles; rowsPerBlock = 8 / wavesPerRow.
// A-frag (16x4 f32): lane l (l16 = l&15, hi = l>>4) holds
//   A[m0+l16][k0 + 2*hi + {0,1}]  -> contiguous float2 load.
// B-frag (4x16 f32): lane holds W[k0 + 2*hi + {0,1}][n0 + l16].
// D/C layout: VGPR v -> row m0 + v + 8*hi, col n0 + l16.
__global__ __launch_bounds__(256)
void wmma_gemm_kernel(const float* __restrict__ A,
                      const float* __restrict__ W,
                      float* __restrict__ C,
                      int nrows, int din, int dout) {
    const int lane = threadIdx.x & 31;
    const int wave = threadIdx.x >> 5;
    const int wavesPerRow  = dout >> 4;
    const int rowsPerBlock = 8 / wavesPerRow;
    const int rowTile = blockIdx.x * rowsPerBlock + wave / wavesPerRow;
    const int colTile = wave % wavesPerRow;
    const int m0 = rowTile * 16;
    const int n0 = colTile * 16;
    if (m0 >= nrows) return;   // wave-uniform: EXEC stays all-ones for WMMA

    const int hi  = lane >> 4;
    const int l16 = lane & 15;

    v8f acc = {0.f, 0.f, 0.f, 0.f, 0.f, 0.f, 0.f, 0.f};

    const float* __restrict__ Arow = A + (size_t)(m0 + l16) * din + 2 * hi;
    const float* __restrict__ Wcol = W + (size_t)(2 * hi) * dout + n0 + l16;

    for (int k0 = 0; k0 < din; k0 += 4) {
        v2f a = *(const v2f*)(Arow + k0);              // 8B-aligned float2
        v2f b;
        const float* bp = Wcol + (size_t)k0 * dout;
        b.x = bp[0];
        b.y = bp[dout];
        acc = __builtin_amdgcn_wmma_f32_16x16x4_f32(
            /*neg_a=*/false, a, /*neg_b=*/false, b,
            /*c_mod=*/(short)0, acc, /*reuse_a=*/false, /*reuse_b=*/false);
    }

    float* __restrict__ Crow = C + (size_t)(m0 + 8 * hi) * dout + n0 + l16;
#pragma unroll
    for (int v = 0; v < 8; ++v)
        Crow[(size_t)v * dout] = acc[v];
}

// ---------------------------------------------------------------- scatter-add
// out[dst] += xw[src] * dinv[src]*dinv[dst]; one edge per wave, D/32 feats/lane.
template <int D>
__global__ __launch_bounds__(256)
void scatter_kernel(const float* __restrict__ xw,
                    const int* __restrict__ srcI,
                    const int* __restrict__ dstI,
                    const float* __restrict__ dinv,
                    float* __restrict__ out) {
    const int lane = threadIdx.x & 31;
    const int wave = threadIdx.x >> 5;
    const long long e = (long long)blockIdx.x * 8 + wave;
    const long long ET = (long long)NEDGES + NNODES;
    if (e >= ET) return;

    int s, d;
    if (e < NEDGES) { s = srcI[e]; d = dstI[e]; }
    else            { s = d = (int)(e - NEDGES); }   // self loop

    const float nrm = dinv[s] * dinv[d];
    constexpr int F = D / 32;                         // 4 (D=128) or 2 (D=64)
    const float* __restrict__ sp = xw  + (size_t)s * D + lane * F;
    float*       __restrict__ dp = out + (size_t)d * D + lane * F;
#pragma unroll
    for (int j = 0; j < F; ++j)
        atomicAdd(&dp[j], sp[j] * nrm);
}

// ---------------------------------------------------------------- bias + relu
__global__ void bias_relu_kernel(float* __restrict__ h,
                                 const float* __restrict__ b,
                                 long long total, int dmask) {
    long long i = (long long)blockIdx.x * blockDim.x + threadIdx.x;
    if (i < total) {
        float v = h[i] + b[(int)i & dmask];
        h[i] = v > 0.0f ? v : 0.0f;
    }
}

// ---------------------------------------------------------------- pooling
// LDS-accumulated segment sum over sorted graph ids, then global flush.
__global__ __launch_bounds__(256)
void pool_kernel(const float* __restrict__ h, const int* __restrict__ batch,
                 float* __restrict__ psum, float* __restrict__ pcnt) {
    __shared__ float ls[NGRAPH * DOUT];
    __shared__ float lc[NGRAPH];
    for (int i = threadIdx.x; i < NGRAPH * DOUT; i += blockDim.x) ls[i] = 0.0f;
    for (int i = threadIdx.x; i < NGRAPH; i += blockDim.x)        lc[i] = 0.0f;
    __syncthreads();

    int n = blockIdx.x * blockDim.x + threadIdx.x;
    if (n < NNODES) {
        int g = batch[n];
        atomicAdd(&lc[g], 1.0f);
        const float* __restrict__ hp = h + (size_t)n * DOUT;
#pragma unroll 4
        for (int f = 0; f < DOUT; ++f)
            atomicAdd(&ls[g * DOUT + f], hp[f]);
    }
    __syncthreads();

    for (int i = threadIdx.x; i < NGRAPH * DOUT; i += blockDim.x) {
        float v = ls[i];
        if (v != 0.0f) atomicAdd(&psum[i], v);
    }
    for (int i = threadIdx.x; i < NGRAPH; i += blockDim.x) {
        float v = lc[i];
        if (v != 0.0f) atomicAdd(&pcnt[i], v);
    }
}

__global__ void finalize_kernel(const float* __restrict__ psum,
                                const float* __restrict__ pcnt,
                                float* __restrict__ out) {
    int i = blockIdx.x * blockDim.x + threadIdx.x;
    if (i < NGRAPH * DOUT)
        out[i] = psum[i] / fmaxf(pcnt[i / DOUT], 1.0f);
}

// ---------------------------------------------------------------- launcher
extern "C" void kernel_launch(void* const* d_in, const int* in_sizes, int n_in,
                              void* d_out, int out_size, void* d_ws, size_t ws_size,
                              hipStream_t stream) {
    const float* x     = (const float*)d_in[0];
    const int*   ei    = (const int*)d_in[1];     // [2, E]
    const int*   batch = (const int*)d_in[2];     // [N]
    const float* W1    = (const float*)d_in[3];
    const float* b1    = (const float*)d_in[4];
    const float* W2    = (const float*)d_in[5];
    const float* b2    = (const float*)d_in[6];
    float*       out   = (float*)d_out;

    const int* srcI = ei;
    const int* dstI = ei + NEDGES;

    // workspace layout (floats)
    float* ws   = (float*)d_ws;
    float* bufA = ws;                              // N*128: xw1, later xw2|agg2
    float* bufB = bufA + (size_t)NNODES * DH;      // N*128: agg1 -> h1
    float* dinv = bufB + (size_t)NNODES * DH;      // N
    float* psum = dinv + NNODES;                   // G*DOUT
    float* pcnt = psum + NGRAPH * DOUT;            // G  (contiguous with psum)
    float* xw2  = bufA;                            // N*64
    float* agg2 = bufA + (size_t)NNODES * DOUT;    // N*64

    const int T = 256;
    const long long ET = (long long)NEDGES + NNODES;
    const int edgeBlocks = (int)((ET + 7) / 8);

    // 1) symmetric degree normalization (self loop included via fill=1)
    fill_kernel<<<(NNODES + T - 1) / T, T, 0, stream>>>(dinv, 1.0f, NNODES);
    degree_kernel<<<(NEDGES + T - 1) / T, T, 0, stream>>>(dstI, dinv);
    dinv_kernel<<<(NNODES + T - 1) / T, T, 0, stream>>>(dinv);

    // 2) layer 1: xw1 = x @ W1   (N/16 row tiles * 8 col tiles / 8 waves)
    wmma_gemm_kernel<<<(NNODES / 16) * (DH / 16) / 8, T, 0, stream>>>(
        x, W1, bufA, NNODES, DIN, DH);

    // 3) agg1 = scatter_add(norm * xw1[src])
    fill_kernel<<<((NNODES * DH) + T - 1) / T, T, 0, stream>>>(bufB, 0.0f, NNODES * DH);
    scatter_kernel<DH><<<edgeBlocks, T, 0, stream>>>(bufA, srcI, dstI, dinv, bufB);

    // 4) h1 = relu(agg1 + b1)
    bias_relu_kernel<<<((NNODES * DH) + T - 1) / T, T, 0, stream>>>(
        bufB, b1, (long long)NNODES * DH, DH - 1);

    // 5) layer 2: xw2 = h1 @ W2
    wmma_gemm_kernel<<<(NNODES / 16) * (DOUT / 16) / 8, T, 0, stream>>>(
        bufB, W2, xw2, NNODES, DH, DOUT);

    // 6) agg2 = scatter_add(norm * xw2[src])
    fill_kernel<<<((NNODES * DOUT) + T - 1) / T, T, 0, stream>>>(agg2, 0.0f, NNODES * DOUT);
    scatter_kernel<DOUT><<<edgeBlocks, T, 0, stream>>>(xw2, srcI, dstI, dinv, agg2);

    // 7) h2 = relu(agg2 + b2)
    bias_relu_kernel<<<((NNODES * DOUT) + T - 1) / T, T, 0, stream>>>(
        agg2, b2, (long long)NNODES * DOUT, DOUT - 1);

    // 8) global mean pool
    fill_kernel<<<(NGRAPH * DOUT + NGRAPH + T - 1) / T, T, 0, stream>>>(
        psum, 0.0f, NGRAPH * DOUT + NGRAPH);
    pool_kernel<<<(NNODES + T - 1) / T, T, 0, stream>>>(agg2, batch, psum, pcnt);
    finalize_kernel<<<(NGRAPH * DOUT + T - 1) / T, T, 0, stream>>>(psum, pcnt, out);
}